// SetAbstraction_27092653703823
// MI455X (gfx1250) — compile-verified
//
#include <hip/hip_runtime.h>
#include <cstdint>
#include <cstddef>

#define NB      16
#define NPTS    4096
#define NFEAT   64
#define NPOINT  1024
#define NSAMPLE 32
#define RADIUS2 0.04f
#define NROWS   (NB * NPOINT * NSAMPLE)   // 524288
#define BN_EPS  1e-5f
#define FLT_BIG 3.4e38f

typedef __attribute__((ext_vector_type(2))) float v2f;
typedef __attribute__((ext_vector_type(8))) float v8f;

// ---------------------------------------------------------------------------
// 1) Farthest point sampling: one block per batch, xyz in LDS, dist in regs.
// ---------------------------------------------------------------------------
__global__ __launch_bounds__(256) void fps_kernel(const float* __restrict__ xyz,
                                                  float* __restrict__ new_xyz) {
  __shared__ float px[NPTS], py[NPTS], pz[NPTS];
  __shared__ float rv[256];
  __shared__ int   ri[256];
  __shared__ int   s_far;

  const int b = blockIdx.x;
  const int t = threadIdx.x;
  const float* base = xyz + (size_t)b * NPTS * 3;
  for (int i = t; i < NPTS; i += 256) {
    px[i] = base[i * 3 + 0];
    py[i] = base[i * 3 + 1];
    pz[i] = base[i * 3 + 2];
  }
  float dist[16];
#pragma unroll
  for (int j = 0; j < 16; ++j) dist[j] = FLT_BIG;
  if (t == 0) s_far = 0;
  __syncthreads();

  for (int s = 0; s < NPOINT; ++s) {
    const int far = s_far;
    if (t == 0) {
      new_xyz[(b * NPOINT + s) * 3 + 0] = px[far];
      new_xyz[(b * NPOINT + s) * 3 + 1] = py[far];
      new_xyz[(b * NPOINT + s) * 3 + 2] = pz[far];
    }
    const float cx = px[far], cy = py[far], cz = pz[far];
    float bv = -1.0f; int bi = 0;
#pragma unroll
    for (int j = 0; j < 16; ++j) {
      const int i = t * 16 + j;
      const float dx = px[i] - cx, dy = py[i] - cy, dz = pz[i] - cz;
      const float d  = dx * dx + dy * dy + dz * dz;
      const float dm = fminf(dist[j], d);
      dist[j] = dm;
      if (dm > bv) { bv = dm; bi = i; }
    }
    rv[t] = bv; ri[t] = bi;
    __syncthreads();
    for (int off = 128; off > 0; off >>= 1) {
      if (t < off) {
        if (rv[t + off] > rv[t] || (rv[t + off] == rv[t] && ri[t + off] < ri[t])) {
          rv[t] = rv[t + off]; ri[t] = ri[t + off];
        }
      }
      __syncthreads();
    }
    if (t == 0) s_far = ri[0];
    __syncthreads();
  }
}

// ---------------------------------------------------------------------------
// 2) Ball query: one wave per query, exact 32-NN by iterative min extraction.
//    Each lane owns a private 128-point slice of the LDS distance array.
// ---------------------------------------------------------------------------
__global__ __launch_bounds__(128) void ballq_kernel(const float* __restrict__ xyz,
                                                    const float* __restrict__ new_xyz,
                                                    int* __restrict__ group_idx) {
  __shared__ float sd[4][NPTS];   // 64 KB
  const int wave = threadIdx.x >> 5;
  const int lane = threadIdx.x & 31;
  const int q = blockIdx.x * 4 + wave;           // [0, NB*NPOINT)
  const int b = q >> 10;
  const float qx = new_xyz[q * 3 + 0];
  const float qy = new_xyz[q * 3 + 1];
  const float qz = new_xyz[q * 3 + 2];
  const float* base = xyz + (size_t)b * NPTS * 3;
  float* d = sd[wave];

  float lminv = FLT_BIG; int lmini = NPTS;
  for (int tt = 0; tt < NPTS / 32; ++tt) {
    const int i = tt * 32 + lane;
    const float dx = base[i * 3 + 0] - qx;
    const float dy = base[i * 3 + 1] - qy;
    const float dz = base[i * 3 + 2] - qz;
    const float dd = dx * dx + dy * dy + dz * dz;
    d[i] = dd;
    if (dd < lminv || (dd == lminv && i < lmini)) { lminv = dd; lmini = i; }
  }

  int   my_idx = 0;
  float my_d   = 0.0f;
  for (int k = 0; k < NSAMPLE; ++k) {
    float v = lminv; int idx = lmini;
    for (int off = 16; off > 0; off >>= 1) {
      const float ov = __shfl_xor(v, off, 32);
      const int   oi = __shfl_xor(idx, off, 32);
      if (ov < v || (ov == v && oi < idx)) { v = ov; idx = oi; }
    }
    if (lane == k) { my_idx = idx; my_d = v; }
    if ((idx & 31) == lane) {           // winner lives in my slice: remove + rescan
      d[idx] = FLT_BIG;
      lminv = FLT_BIG; lmini = NPTS;
      for (int tt = 0; tt < NPTS / 32; ++tt) {
        const int i = tt * 32 + lane;
        const float dd = d[i];
        if (dd < lminv) { lminv = dd; lmini = i; }
      }
    }
  }
  const int first = __shfl(my_idx, 0, 32);
  group_idx[(size_t)q * NSAMPLE + lane] = (my_d > RADIUS2) ? first : my_idx;
}

// ---------------------------------------------------------------------------
// 3) GEMM1: fused gather + (524288 x 68) * (68 x 64) via v_wmma_f32_16x16x4_f32
// ---------------------------------------------------------------------------
__global__ __launch_bounds__(128) void gemm1_kernel(const float* __restrict__ xyz,
                                                    const float* __restrict__ feat,
                                                    const float* __restrict__ new_xyz,
                                                    const int* __restrict__ group_idx,
                                                    const float* __restrict__ w1,
                                                    float* __restrict__ y1) {
  __shared__ float xs[16][72];   // 68 valid cols, padded stride
  __shared__ float wt[68][64];   // wt[k][n] = w1[n][k], k>=67 zero
  const int t = threadIdx.x;
  const int lane = t & 31, wv = t >> 5;

  for (int e = t; e < 68 * 64; e += 128) {
    const int k = e >> 6, n = e & 63;
    wt[k][n] = (k < 67) ? w1[n * 67 + k] : 0.0f;
  }
  const int r0 = blockIdx.x * 16;
  for (int e = t; e < 16 * 68; e += 128) {
    const int row = e / 68, col = e % 68;
    const int r = r0 + row;
    const int q = r >> 5;                 // b*NPOINT + s
    const int b = q >> 10;
    const int idx = group_idx[r];
    float v;
    if (col < 3)       v = xyz[((size_t)b * NPTS + idx) * 3 + col] - new_xyz[q * 3 + col];
    else if (col < 67) v = feat[((size_t)b * NPTS + idx) * NFEAT + (col - 3)];
    else               v = 0.0f;
    xs[row][col] = v;
  }
  __syncthreads();

  const int hi = lane >> 4, nl = lane & 15;
  const int n0 = wv * 16;
  v8f acc = {};
#pragma unroll
  for (int k0 = 0; k0 < 68; k0 += 4) {
    const int ka = k0 + 2 * hi;
    v2f a  = { xs[nl][ka], xs[nl][ka + 1] };
    v2f bb = { wt[ka][n0 + nl], wt[ka + 1][n0 + nl] };
    acc = __builtin_amdgcn_wmma_f32_16x16x4_f32(false, a, false, bb, (short)0, acc,
                                                false, false);
  }
#pragma unroll
  for (int v = 0; v < 8; ++v) {
    const int row = v + 8 * hi;
    y1[(size_t)(r0 + row) * 64 + n0 + nl] = acc[v];
  }
}

// ---------------------------------------------------------------------------
// 4) Deterministic per-channel sum/sumsq: two fixed-order stages (no atomics).
// ---------------------------------------------------------------------------
template <int C>
__global__ __launch_bounds__(256) void stats_partial_kernel(const float* __restrict__ y,
                                                            float* __restrict__ partial,
                                                            int rtot) {
  constexpr int RG = 256 / C;
  __shared__ float ls[256], lq[256];
  const int t = threadIdx.x;
  const int ch = t & (C - 1);
  const int rg = t / C;
  float s = 0.0f, sq = 0.0f;
  for (int r = blockIdx.x * RG + rg; r < rtot; r += gridDim.x * RG) {
    const float v = y[(size_t)r * C + ch];
    s += v; sq += v * v;
  }
  ls[t] = s; lq[t] = sq;
  __syncthreads();
  for (int off = 128; off >= C; off >>= 1) {
    if (t < off) { ls[t] += ls[t + off]; lq[t] += lq[t + off]; }
    __syncthreads();
  }
  if (t < C) {
    partial[(size_t)blockIdx.x * 2 * C + t]     = ls[t];
    partial[(size_t)blockIdx.x * 2 * C + C + t] = lq[t];
  }
}

template <int C>
__global__ __launch_bounds__(2 * C) void stats_reduce_kernel(const float* __restrict__ partial,
                                                             float* __restrict__ sums,
                                                             int nblk) {
  const int t = threadIdx.x;   // 2*C threads
  float s = 0.0f;
  for (int i = 0; i < nblk; ++i) s += partial[(size_t)i * 2 * C + t];
  sums[t] = s;
}

// ---------------------------------------------------------------------------
// 5) GEMM with fused input BN+ReLU (applied while staging to LDS).
// ---------------------------------------------------------------------------
template <int COUT>
__global__ __launch_bounds__(2 * COUT) void gemm_bn_kernel(const float* __restrict__ yin,
                                                           const float* __restrict__ stats,
                                                           const float* __restrict__ gamma,
                                                           const float* __restrict__ beta,
                                                           const float* __restrict__ w,
                                                           float* __restrict__ yout) {
  constexpr int NWAVE = COUT / 16;
  constexpr int NTHR  = 32 * NWAVE;
  __shared__ float xs[16][68];
  __shared__ float wt[64][COUT];
  __shared__ float sc[64], sh[64];
  const int t = threadIdx.x;
  const int lane = t & 31, wv = t >> 5;

  if (t < 64) {
    const float mean = stats[t] * (1.0f / NROWS);
    const float var  = stats[64 + t] * (1.0f / NROWS) - mean * mean;
    const float s    = rsqrtf(var + BN_EPS) * gamma[t];
    sc[t] = s;
    sh[t] = beta[t] - mean * s;
  }
  for (int e = t; e < 64 * COUT; e += NTHR) {
    const int k = e / COUT, n = e % COUT;
    wt[k][n] = w[n * 64 + k];
  }
  __syncthreads();

  const size_t r0 = (size_t)blockIdx.x * 16;
  for (int e = t; e < 16 * 64; e += NTHR) {
    const int row = e >> 6, col = e & 63;
    const float v = yin[(r0 + row) * 64 + col] * sc[col] + sh[col];
    xs[row][col] = fmaxf(v, 0.0f);
  }
  __syncthreads();

  const int hi = lane >> 4, nl = lane & 15;
  const int n0 = wv * 16;
  v8f acc = {};
#pragma unroll
  for (int k0 = 0; k0 < 64; k0 += 4) {
    const int ka = k0 + 2 * hi;
    v2f a  = { xs[nl][ka], xs[nl][ka + 1] };
    v2f bb = { wt[ka][n0 + nl], wt[ka + 1][n0 + nl] };
    acc = __builtin_amdgcn_wmma_f32_16x16x4_f32(false, a, false, bb, (short)0, acc,
                                                false, false);
  }
#pragma unroll
  for (int v = 0; v < 8; ++v) {
    const int row = v + 8 * hi;
    yout[(r0 + row) * COUT + n0 + nl] = acc[v];
  }
}

// ---------------------------------------------------------------------------
// 6) BN3 + ReLU + max over the 32 samples (relu commutes with max).
// ---------------------------------------------------------------------------
__global__ __launch_bounds__(256) void bn_max_kernel(const float* __restrict__ y3,
                                                     const float* __restrict__ stats,
                                                     const float* __restrict__ gamma,
                                                     const float* __restrict__ beta,
                                                     float* __restrict__ out_feat) {
  const int i = blockIdx.x * 256 + threadIdx.x;    // over NB*NPOINT*128
  const int ch = i & 127;
  const int q  = i >> 7;
  const float mean  = stats[ch] * (1.0f / NROWS);
  const float var   = stats[128 + ch] * (1.0f / NROWS) - mean * mean;
  const float scale = rsqrtf(var + BN_EPS) * gamma[ch];
  const float shift = beta[ch] - mean * scale;
  const float* p = y3 + (size_t)q * NSAMPLE * 128 + ch;
  float m = -FLT_BIG;
#pragma unroll 4
  for (int kk = 0; kk < NSAMPLE; ++kk) m = fmaxf(m, p[(size_t)kk * 128] * scale + shift);
  out_feat[i] = fmaxf(m, 0.0f);
}

// ---------------------------------------------------------------------------
extern "C" void kernel_launch(void* const* d_in, const int* in_sizes, int n_in,
                              void* d_out, int out_size, void* d_ws, size_t ws_size,
                              hipStream_t stream) {
  (void)in_sizes; (void)n_in; (void)out_size; (void)ws_size;
  const float* xyz  = (const float*)d_in[0];
  const float* feat = (const float*)d_in[1];
  const float* w1 = (const float*)d_in[2];
  const float* g1 = (const float*)d_in[3];
  const float* b1 = (const float*)d_in[4];
  const float* w2 = (const float*)d_in[5];
  const float* g2 = (const float*)d_in[6];
  const float* b2 = (const float*)d_in[7];
  const float* w3 = (const float*)d_in[8];
  const float* g3 = (const float*)d_in[9];
  const float* b3 = (const float*)d_in[10];

  float* out      = (float*)d_out;
  float* new_xyz  = out;                       // NB*NPOINT*3
  float* out_feat = out + NB * NPOINT * 3;     // NB*NPOINT*128

  char* ws = (char*)d_ws;
  size_t off = 0;
  auto alloc = [&](size_t bytes) -> void* {
    void* p = ws + off;
    off = (off + bytes + 255) & ~(size_t)255;
    return p;
  };
  int*   group_idx = (int*)  alloc((size_t)NROWS * sizeof(int));
  float* y1        = (float*)alloc((size_t)NROWS * 64  * sizeof(float));
  float* y2        = (float*)alloc((size_t)NROWS * 64  * sizeof(float));
  float* y3        = (float*)alloc((size_t)NROWS * 128 * sizeof(float));
  float* stats1    = (float*)alloc(128 * sizeof(float));   // sum[64], sumsq[64]
  float* stats2    = (float*)alloc(128 * sizeof(float));
  float* stats3    = (float*)alloc(256 * sizeof(float));   // sum[128], sumsq[128]
  constexpr int SBLK = 512;
  float* partial   = (float*)alloc((size_t)SBLK * 256 * sizeof(float));

  fps_kernel<<<NB, 256, 0, stream>>>(xyz, new_xyz);
  ballq_kernel<<<(NB * NPOINT) / 4, 128, 0, stream>>>(xyz, new_xyz, group_idx);

  gemm1_kernel<<<NROWS / 16, 128, 0, stream>>>(xyz, feat, new_xyz, group_idx, w1, y1);
  stats_partial_kernel<64><<<SBLK, 256, 0, stream>>>(y1, partial, NROWS);
  stats_reduce_kernel<64><<<1, 128, 0, stream>>>(partial, stats1, SBLK);

  gemm_bn_kernel<64><<<NROWS / 16, 128, 0, stream>>>(y1, stats1, g1, b1, w2, y2);
  stats_partial_kernel<64><<<SBLK, 256, 0, stream>>>(y2, partial, NROWS);
  stats_reduce_kernel<64><<<1, 128, 0, stream>>>(partial, stats2, SBLK);

  gemm_bn_kernel<128><<<NROWS / 16, 256, 0, stream>>>(y2, stats2, g2, b2, w3, y3);
  stats_partial_kernel<128><<<SBLK, 256, 0, stream>>>(y3, partial, NROWS);
  stats_reduce_kernel<128><<<1, 256, 0, stream>>>(partial, stats3, SBLK);

  bn_max_kernel<<<(NB * NPOINT * 128) / 256, 256, 0, stream>>>(y3, stats3, g3, b3, out_feat);
}